// SWTInverse_50259707298481
// MI455X (gfx1250) — compile-verified
//
#include <hip/hip_runtime.h>

// Inverse SWT (Haar-like 2x2 circular stencil), B=8, C=64, H=W=256, fp32.
// out[h,w] = 0.25 * sum_band( w00*X[h-1,w-1] + w01*X[h-1,w] + w10*X[h,w-1] + w11*X[h,w] )
// Bandwidth-bound: ~671 MB HBM traffic -> ~29us floor at 23.3 TB/s.
// Strategy: B128 coalesced loads, per-thread 4-row reuse, NT streaming stores.

#define PLANE_ELEMS 65536   // 256*256
#define ROW_ELEMS   256

// Native clang vector type (required by __builtin_nontemporal_store; also
// guarantees b128 vector memory ops).
typedef __attribute__((ext_vector_type(4))) float f32x4;

__global__ __launch_bounds__(256) void swt_inverse_kernel(
    const float* __restrict__ LL, const float* __restrict__ LH,
    const float* __restrict__ HL, const float* __restrict__ HH,
    const float* __restrict__ lor, const float* __restrict__ hir,
    float* __restrict__ out)
{
    const int tid   = blockIdx.x * blockDim.x + threadIdx.x;
    const int w4    = tid & 63;          // which float4 column (0..63)
    const int h4    = (tid >> 6) & 63;   // which 4-row tile (0..63)
    const int plane = tid >> 12;         // b*C + c  (0..511)

    const int c    = w4 << 2;                               // first output column
    const int cm   = (c == 0) ? (ROW_ELEMS - 1) : (c - 1);  // wrapped left-halo col
    const int h0   = h4 << 2;                               // first output row
    const int rtop = (h0 == 0) ? (ROW_ELEMS - 1) : (h0 - 1); // wrapped top halo row

    const long base = (long)plane * PLANE_ELEMS;

    // Uniform 2-tap kernels -> SGPRs via scalar loads.
    const float l0 = lor[0], l1 = lor[1];
    const float g0 = hir[0], g1 = hir[1];

    // Per-band combined weights (0.25 folded in).
    // band order: LL(krow=lor,kcol=lor) LH(lor,hir) HL(hir,lor) HH(hir,hir)
    float w00[4], w01[4], w10[4], w11[4];
    {
        const float kr0[4] = {l0, l0, g0, g0};
        const float kr1[4] = {l1, l1, g1, g1};
        const float kc0[4] = {l0, g0, l0, g0};
        const float kc1[4] = {l1, g1, l1, g1};
#pragma unroll
        for (int b = 0; b < 4; ++b) {
            w00[b] = 0.25f * kr0[b] * kc0[b];  // (top, left)
            w01[b] = 0.25f * kr0[b] * kc1[b];  // (top, same)
            w10[b] = 0.25f * kr1[b] * kc0[b];  // (bot, left)
            w11[b] = 0.25f * kr1[b] * kc1[b];  // (bot, same)
        }
    }

    const float* __restrict__ X[4] = {LL + base, LH + base, HL + base, HH + base};

    // Prime the top halo row (prev = row h0-1, circular).
    f32x4 pv[4];
    float ps[4];
#pragma unroll
    for (int b = 0; b < 4; ++b) {
        const float* row = X[b] + rtop * ROW_ELEMS;
        pv[b] = *reinterpret_cast<const f32x4*>(row + c);   // global_load_b128
        ps[b] = row[cm];                                    // left-halo scalar (cache hit)
    }

    float* __restrict__ orow = out + base + (long)h0 * ROW_ELEMS + c;

#pragma unroll
    for (int k = 0; k < 4; ++k) {
        f32x4 cv[4];
        float cs[4];
#pragma unroll
        for (int b = 0; b < 4; ++b) {
            const float* row = X[b] + (h0 + k) * ROW_ELEMS;
            cv[b] = *reinterpret_cast<const f32x4*>(row + c);
            cs[b] = row[cm];
        }

        f32x4 acc = (f32x4)(0.0f);
#pragma unroll
        for (int b = 0; b < 4; ++b) {
            // shifted-prev = (ps, pv.x, pv.y, pv.z); shifted-cur = (cs, cv.x, cv.y, cv.z)
            acc.x = fmaf(w00[b], ps[b],   acc.x);
            acc.y = fmaf(w00[b], pv[b].x, acc.y);
            acc.z = fmaf(w00[b], pv[b].y, acc.z);
            acc.w = fmaf(w00[b], pv[b].z, acc.w);

            acc.x = fmaf(w01[b], pv[b].x, acc.x);
            acc.y = fmaf(w01[b], pv[b].y, acc.y);
            acc.z = fmaf(w01[b], pv[b].z, acc.z);
            acc.w = fmaf(w01[b], pv[b].w, acc.w);

            acc.x = fmaf(w10[b], cs[b],   acc.x);
            acc.y = fmaf(w10[b], cv[b].x, acc.y);
            acc.z = fmaf(w10[b], cv[b].y, acc.z);
            acc.w = fmaf(w10[b], cv[b].z, acc.w);

            acc.x = fmaf(w11[b], cv[b].x, acc.x);
            acc.y = fmaf(w11[b], cv[b].y, acc.y);
            acc.z = fmaf(w11[b], cv[b].z, acc.z);
            acc.w = fmaf(w11[b], cv[b].w, acc.w);

            pv[b] = cv[b];
            ps[b] = cs[b];
        }

        // Write-once output: non-temporal streaming store (gfx1250 TH hint),
        // keeps cache lines free for the input halo reuse across threads.
        __builtin_nontemporal_store(acc, reinterpret_cast<f32x4*>(orow + k * ROW_ELEMS));
    }
}

extern "C" void kernel_launch(void* const* d_in, const int* in_sizes, int n_in,
                              void* d_out, int out_size, void* d_ws, size_t ws_size,
                              hipStream_t stream) {
    (void)in_sizes; (void)n_in; (void)d_ws; (void)ws_size; (void)out_size;

    const float* LL  = (const float*)d_in[0];
    const float* LH  = (const float*)d_in[1];
    const float* HL  = (const float*)d_in[2];
    const float* HH  = (const float*)d_in[3];
    const float* lor = (const float*)d_in[4];
    const float* hir = (const float*)d_in[5];
    float* out = (float*)d_out;

    // B*C = 512 planes, 64 row-tiles * 64 col-quads each -> 2,097,152 threads.
    const int total_threads = 512 * 64 * 64;
    const int block = 256;
    const int grid  = total_threads / block;  // 8192

    swt_inverse_kernel<<<grid, block, 0, stream>>>(LL, LH, HL, HH, lor, hir, out);
}